// CTCLossLayer_49598282334769
// MI455X (gfx1250) — compile-verified
//
#include <hip/hip_runtime.h>

// CTC batch cost (Keras semantics), B=256, T=512, C=128, L=128, on gfx1250.
// One block per batch element; 288 threads (9 wave32) cover S=2L+1=257 cells,
// one alpha cell per thread held in a REGISTER. Neighbor cells alpha[s-1],
// alpha[s-2] come from wave32 shuffles; only the top two lanes of each wave
// export their value to LDS (parity double-buffered) for the next wave.
// => exactly ONE barrier per timestep.
// Probability rows (512B each) stream through a 4-deep LDS ring via
// GLOBAL_LOAD_ASYNC_TO_LDS_B128 (ASYNCcnt), keeping up to 3 rows in flight.

#define CTC_NEG (-1e30f)
#define CTC_EPS (1e-7f)

constexpr int kB = 256;
constexpr int kT = 512;
constexpr int kC = 128;
constexpr int kL = 128;
constexpr int kS = 2 * kL + 1;   // 257
constexpr int kBlank = kC - 1;   // 127
constexpr int kWaves = 9;
constexpr int kBlock = kWaves * 32;  // 288
constexpr int kRing = 4;             // async row ring depth

// 512B async row copy: 32 lanes x 16B. GVS addressing:
//   global = SGPR64 base + VGPR32 byte offset.
// vdst = LDS byte-address VGPR, vaddr = 32-bit global offset VGPR,
// saddr = SGPR pair base. Tracked by ASYNCcnt (in-order completion).
__device__ __forceinline__ void async_row_load(unsigned long long sbase,
                                               int t, unsigned lds_byte_addr,
                                               int lane) {
  unsigned voff = (unsigned)(t * (kC * 4) + lane * 16);
  asm volatile("global_load_async_to_lds_b128 %0, %1, %2"
               :: "v"(lds_byte_addr), "v"(voff), "s"(sbase)
               : "memory");
}

__global__ __launch_bounds__(kBlock) void ctc_loss_kernel(
    const int* __restrict__ labels,        // [B, L]
    const float* __restrict__ y_pred,      // [B, T, C] softmax probs
    const int* __restrict__ input_length,  // [B, 1]
    const int* __restrict__ label_length,  // [B, 1]
    float* __restrict__ out)               // [B, 1]
{
  __shared__ float rows[kRing][kC];     // async row ring (4 x 512B)
  __shared__ float bnd[2][kWaves][2];   // per-wave boundary cells, by t-parity
  __shared__ float afin[kS];            // final alpha dump

  const int b    = blockIdx.x;
  const int tid  = threadIdx.x;
  const int s    = tid;
  const int wv   = tid >> 5;
  const int lane = tid & 31;

  const float* ybase = y_pred + (size_t)b * kT * kC;
  const unsigned long long sbase = (unsigned long long)(uintptr_t)ybase;

  int il = input_length[b];
  il = il < 1 ? 1 : (il > kT ? kT : il);
  const int t_end = il - 1;  // reference: t_idx = clip(input_length,1,T)-1

  // ext[s] = blank (even s) or labels[(s-1)/2] (odd s);
  // allow_skip[s] = odd s && ext[s] != ext[s-2] (labels never equal blank).
  int  my_ext   = kBlank;
  bool my_allow = false;
  if (s < kS && (s & 1)) {
    const int i = s >> 1;
    my_ext = labels[b * kL + i];
    my_allow = (i >= 1) ? (my_ext != labels[b * kL + i - 1]) : true;
  }

  // LDS byte addresses of this lane's 16B slot in each ring buffer
  // (low 32 bits of a generic LDS pointer are the LDS byte offset).
  unsigned ldsrow[kRing] = {0, 0, 0, 0};
  if (tid < 32) {
    for (int k = 0; k < kRing; ++k)
      ldsrow[k] = (unsigned)(uintptr_t)&rows[k][lane * 4];
    // Prologue: fill the ring (rows 0..min(3,t_end)), drain once.
    for (int k = 0; k < kRing; ++k)
      if (k <= t_end) async_row_load(sbase, k, ldsrow[k], lane);
    asm volatile("s_wait_asynccnt 0" ::: "memory");
  }
  __syncthreads();

  // alpha0: only s=0 (blank) and s=1 (first label) live; rest NEG.
  float aOld = (s < 2) ? __builtin_logf(rows[0][my_ext] + CTC_EPS) : CTC_NEG;
  if (lane >= 30) bnd[0][wv][lane - 30] = aOld;

  for (int t = 1; t <= t_end; ++t) {
    // Row t landed when at most the 2 younger copies are still outstanding
    // (in-order completion on ASYNCcnt).
    if (tid < 32) {
      if (t + 2 <= t_end)      asm volatile("s_wait_asynccnt 2" ::: "memory");
      else if (t + 1 <= t_end) asm volatile("s_wait_asynccnt 1" ::: "memory");
      else                     asm volatile("s_wait_asynccnt 0" ::: "memory");
    }
    __syncthreads();  // row t ready; bnd[(t-1)&1] complete; body t-1 retired

    // Refill the slot last read in body t-1 (safe: barrier above retired it).
    if (tid < 32 && t + 3 <= t_end)
      async_row_load(sbase, t + 3, ldsrow[(t + 3) & (kRing - 1)], lane);

    // Neighbors via wave shuffles; cross-wave cells from LDS boundary slots.
    float a1 = __shfl_up(aOld, 1);
    float a2 = __shfl_up(aOld, 2);
    if (lane == 0) {
      a1 = (wv > 0) ? bnd[(t - 1) & 1][wv - 1][1] : CTC_NEG;
      a2 = (wv > 0) ? bnd[(t - 1) & 1][wv - 1][0] : CTC_NEG;
    } else if (lane == 1) {
      a2 = (wv > 0) ? bnd[(t - 1) & 1][wv - 1][1] : CTC_NEG;
    }
    if (!my_allow) a2 = CTC_NEG;  // even s / repeated label / s<2

    const float m   = fmaxf(aOld, fmaxf(a1, a2));
    const float lse = m + __builtin_logf(__builtin_expf(aOld - m) +
                                         __builtin_expf(a1 - m) +
                                         __builtin_expf(a2 - m));
    const float lp  = __builtin_logf(rows[t & (kRing - 1)][my_ext] + CTC_EPS);
    const float aNew = lp + lse;

    if (lane >= 30) bnd[t & 1][wv][lane - 30] = aNew;
    aOld = aNew;
  }

  // Gather the two terminal cells: alpha[2*ll] and alpha[2*ll - 1].
  if (s < kS) afin[s] = aOld;
  __syncthreads();
  if (tid == 0) {
    const int sL = 2 * label_length[b];  // label_length in [1, L]
    const float aE = afin[sL];
    const float aP = afin[sL - 1];
    const float m  = fmaxf(aE, aP);
    out[b] = -(m + __builtin_logf(__builtin_expf(aE - m) +
                                  __builtin_expf(aP - m)));
  }
}

extern "C" void kernel_launch(void* const* d_in, const int* in_sizes, int n_in,
                              void* d_out, int out_size, void* d_ws, size_t ws_size,
                              hipStream_t stream) {
  (void)in_sizes; (void)n_in; (void)out_size; (void)d_ws; (void)ws_size;
  const int*   labels       = (const int*)d_in[0];
  const float* y_pred       = (const float*)d_in[1];
  const int*   input_length = (const int*)d_in[2];
  const int*   label_length = (const int*)d_in[3];
  float*       out          = (float*)d_out;
  ctc_loss_kernel<<<dim3(kB), dim3(kBlock), 0, stream>>>(
      labels, y_pred, input_length, label_length, out);
}